// WaterNetModel2_26414048870931
// MI455X (gfx1250) — compile-verified
//
#include <hip/hip_runtime.h>
#include <stdint.h>

#define NT   2000
#define NS   2048
#define NH   16
#define SPB  32                 // sites per block
#define TCH  32                 // timesteps per chunk
#define NCHUNK ((NT + TCH - 1) / TCH)   // 63 (last chunk = 16 rows)
#define BLOCK 512               // 16 wave32s, 32 sites * 16 hidden

#ifndef __has_builtin
#define __has_builtin(x) 0
#endif

#if __has_builtin(__builtin_amdgcn_tensor_load_to_lds)
#define USE_TDM 1
#else
#define USE_TDM 0
#endif

typedef unsigned int u32x4 __attribute__((ext_vector_type(4)));
typedef int          i32x8 __attribute__((ext_vector_type(8)));
typedef int          i32x4 __attribute__((ext_vector_type(4)));

__device__ __forceinline__ float sigm(float x) { return 1.0f / (1.0f + expf(-x)); }

__device__ __forceinline__ void wait_tensor0() {
#if __has_builtin(__builtin_amdgcn_s_wait_tensorcnt)
  __builtin_amdgcn_s_wait_tensorcnt(0);
#else
  asm volatile("s_wait_tensorcnt 0x0" ::: "memory");
#endif
}

#if USE_TDM
// Issue one TDM 2D-tile load: tile_dim0 = SPB contiguous f32 (128B rows),
// tile_dim1 = rows, row stride = NS elements. D# packed per CDNA5 ISA 8.3/8.4.
// This toolchain exposes the 6-arg builtin:
//   (uint32x4 g0, int32x8 g1, int32x4 g2, int32x4 g3, int32x8 pad, int cpol)
__device__ __forceinline__ void tdm_load_tile(unsigned lds_addr,
                                              const float* gaddr, int rows) {
  unsigned long long ga = (unsigned long long)(uintptr_t)gaddr;
  const unsigned td0 = (unsigned)SPB;        // tensor_dim0 == tile_dim0 (no OOB)
  const unsigned td1 = (unsigned)rows;       // tensor_dim1 == tile_dim1
  u32x4 g0;
  g0[0] = 1u;                                           // count=1, user mode
  g0[1] = lds_addr;                                     // LDS byte address
  g0[2] = (unsigned)ga;                                 // global addr [31:0]
  g0[3] = (unsigned)((ga >> 32) & 0x01FFFFFFu) | (2u << 30);  // [56:32], type=2
  i32x8 g1;
  g1[0] = (int)(2u << 16);                              // data_size = 4 bytes
  g1[1] = (int)((td0 & 0xFFFFu) << 16);                 // tensor_dim0[15:0]
  g1[2] = (int)((td0 >> 16) | ((td1 & 0xFFFFu) << 16)); // d0 hi16 | d1 lo16
  g1[3] = (int)((td1 >> 16) | ((unsigned)SPB << 16));   // d1 hi16 | tile_dim0
  g1[4] = (int)(td1 & 0xFFFFu);                         // tile_dim1 (tile_dim2=0)
  g1[5] = (int)(unsigned)NS;                            // tensor_dim0_stride lo32
  g1[6] = 0;                                            // stride hi | d1stride lo
  g1[7] = 0;
  i32x4 z4 = {0, 0, 0, 0};
  i32x8 z8 = {0, 0, 0, 0, 0, 0, 0, 0};
  __builtin_amdgcn_tensor_load_to_lds(g0, g1, z4, z4, z8, 0);
}
#endif

__global__ __launch_bounds__(BLOCK, 1) void waternet_scan_kernel(
    const float* __restrict__ P, const float* __restrict__ T,
    const float* __restrict__ E, const float* __restrict__ w_o,
    const float* __restrict__ wF, const float* __restrict__ wG,
    const float* __restrict__ wl, const float* __restrict__ we,
    const float* __restrict__ wk, const float* __restrict__ ws,
    float* __restrict__ out) {
  __shared__ float sb[2][3][TCH][SPB];   // 24 KB double buffer for P/T/E tiles

  const int tid  = threadIdx.x;
  const int hid  = tid & (NH - 1);       // hidden index within half-wave
  const int sl   = tid >> 4;             // local site 0..31

  // ---- one-time per-lane parameter setup (uniform scalar loads) ----
  float mx = w_o[0];
#pragma unroll
  for (int j = 1; j < NH; ++j) mx = fmaxf(mx, w_o[j]);
  float ssum = 0.0f;
#pragma unroll
  for (int j = 0; j < NH; ++j) ssum += expf(w_o[j] - mx);
  const float a_h  = expf(w_o[hid] - mx) / ssum;   // softmax mixing weight
  const float melt = expf(wF[hid]) + 1.0f;
  const float cap  = expf(2.0f * wl[hid]);
  const float k_we = sigm(we[hid]);
  const float k_wk = sigm(wk[hid]);
  const float k_ws = sigm(ws[hid]);
  const float oms  = 1.0f - k_ws;
  const float k_wg = sigm(wG[hid]);

  // ---- output pointers (tuple order: Q, Fh, Hh, Gh) ----
  float* Q  = out;
  float* Fh = out + (size_t)NT * NS;
  float* Hh = Fh + (size_t)NT * NS * NH;
  float* Gh = Hh + (size_t)NT * NS * NH;

  float stF = 0.0f, stH = 0.0f, stG = 0.0f;

  // chunk loader: TDM path issued by wave 0 only; fallback is cooperative
  auto load_chunk = [&](int c, int buf) {
#if USE_TDM
    if (tid < 32) {
      const int t0   = c * TCH;
      const int rows = (NT - t0 < TCH) ? (NT - t0) : TCH;
      const size_t go = (size_t)t0 * NS + (size_t)blockIdx.x * SPB;
      const unsigned ldsb = (unsigned)(uintptr_t)(void*)&sb[buf][0][0][0];
      tdm_load_tile(ldsb + 0u * TCH * SPB * 4u, P + go, rows);
      tdm_load_tile(ldsb + 1u * TCH * SPB * 4u, T + go, rows);
      tdm_load_tile(ldsb + 2u * TCH * SPB * 4u, E + go, rows);
    }
#else
    const int t0 = c * TCH;
    for (int i = tid; i < 3 * TCH * SPB; i += BLOCK) {
      const int s   = i & (SPB - 1);
      const int tt  = (i >> 5) & (TCH - 1);
      const int arr = i >> 10;
      int t = t0 + tt;
      if (t > NT - 1) t = NT - 1;            // clamped over-read, never consumed
      const float* src = (arr == 0) ? P : ((arr == 1) ? T : E);
      sb[buf][arr][tt][s] = src[(size_t)t * NS + (size_t)blockIdx.x * SPB + s];
    }
#endif
  };

  load_chunk(0, 0);                          // prime the pipeline

  for (int c = 0; c < NCHUNK; ++c) {
    const int t0   = c * TCH;
    const int clen = (NT - t0 < TCH) ? (NT - t0) : TCH;
    const int buf  = c & 1;

#if USE_TDM
    if (tid < 32) wait_tensor0();            // chunk c resident in LDS
#endif
    __syncthreads();
    if (c + 1 < NCHUNK) load_chunk(c + 1, buf ^ 1);   // overlap next tile

    const size_t base = (size_t)t0 * NS + (size_t)blockIdx.x * SPB;
    float* fb = Fh + base * NH + tid;
    float* hb = Hh + base * NH + tid;
    float* gb = Gh + base * NH + tid;
    float* qb = Q + base + sl;

#pragma unroll 4
    for (int tt = 0; tt < clen; ++tt) {
      const float Pk = sb[buf][0][tt][sl];   // LDS broadcast within half-wave
      const float Tk = sb[buf][1][tt][sl];
      const float Ek = sb[buf][2][tt][sl];

      // SnowBucket
      const float sm = fmaxf(Tk, 0.0f) * melt;
      const float mm = fminf(sm, stF);
      stF = stF - mm + (Tk < 0.0f ? Pk : 0.0f);
      const float x = (Tk > 0.0f ? Pk : 0.0f) + mm;
      // SoilBucket
      const float hn = stH + x;
      const float h1 = fmaxf(hn - cap, 0.0f);
      const float q1 = fmaxf(h1 - Ek * k_we, 0.0f);
      const float h2 = hn - h1;
      const float q2 = h2 * k_wk;
      stH = h2 - q2;
      const float q2a = q2 * k_ws;
      const float q2b = q2 * oms;
      // LinearBucket
      const float q3 = (q2b + stG) * k_wg;
      stG = stG - q3 + q2b;

      // coalesced 128B/wave state stores
      const unsigned eo = (unsigned)tt * (NS * NH);
      fb[eo] = stF;
      hb[eo] = stH;
      gb[eo] = stG;

      // Q: weighted 16-lane tree reduction inside each half-wave (wave32)
      float y = (q1 + q2a + q3) * a_h;
      y += __shfl_xor(y, 1, 32);
      y += __shfl_xor(y, 2, 32);
      y += __shfl_xor(y, 4, 32);
      y += __shfl_xor(y, 8, 32);
      if (hid == 0) qb[(unsigned)tt * NS] = y;
    }
    __syncthreads();                          // buffer safe to overwrite
  }
}

extern "C" void kernel_launch(void* const* d_in, const int* in_sizes, int n_in,
                              void* d_out, int out_size, void* d_ws,
                              size_t ws_size, hipStream_t stream) {
  const float* P   = (const float*)d_in[0];
  const float* T   = (const float*)d_in[1];
  const float* E   = (const float*)d_in[2];
  const float* w_o = (const float*)d_in[3];
  const float* wF  = (const float*)d_in[4];
  const float* wG  = (const float*)d_in[5];
  const float* wl  = (const float*)d_in[6];
  const float* we  = (const float*)d_in[7];
  const float* wk  = (const float*)d_in[8];
  const float* wsp = (const float*)d_in[9];
  float* out = (float*)d_out;
  (void)in_sizes; (void)n_in; (void)out_size; (void)d_ws; (void)ws_size;

  dim3 grid(NS / SPB);     // 64 blocks * 16 waves = 1024 wave32s (max scan parallelism)
  dim3 block(BLOCK);
  waternet_scan_kernel<<<grid, block, 0, stream>>>(P, T, E, w_o, wF, wG, wl,
                                                   we, wk, wsp, out);
}